// ModelNew_60713657696911
// MI455X (gfx1250) — compile-verified
//
#include <hip/hip_runtime.h>
#include <stdint.h>

// Problem shape (fixed by the reference file)
#define N_ROWS 8192
#define TWO_H  8192
#define HALF   4096

#define TPB    256
#define VEC    4
#define CHUNK  (TPB * VEC)      // 1024 columns per chunk
#define NCHUNK (HALF / CHUNK)   // 4 chunks per row

typedef __attribute__((ext_vector_type(4))) float v4f;
typedef __attribute__((ext_vector_type(4))) int   v4i;

__device__ __forceinline__ float gate_quant(float xi_a, float xi_b,
                                            float wa, float wb, float as,
                                            float ba, float bb,
                                            float qs, float qo) {
    // deq = x * weight_scale * activate_scale + bias
    float a = fmaf(xi_a * wa, as, ba);
    float b = fmaf(xi_b * wb, as, bb);
    // silu(b) = b * sigmoid(b); fast path: v_exp_f32 + v_rcp_f32
    float e   = __expf(-b);
    float sig = __builtin_amdgcn_rcpf(1.0f + e);
    float g   = a * b * sig;
    float q   = fmaf(g, qs, qo);
    q = rintf(q);                               // round-half-even == jnp.round
    q = fminf(fmaxf(q, -128.0f), 127.0f);       // clip after round (ref order)
    return q;
}

__global__ __launch_bounds__(TPB) void swiglu_dequant_quant_kernel(
    const int*   __restrict__ x,        // (N, 2H) int32
    const float* __restrict__ wscale,   // (2H,)
    const float* __restrict__ ascale,   // (N,)
    const float* __restrict__ bias,     // (2H,)
    const float* __restrict__ qscale,   // (H,)
    const float* __restrict__ qoffset,  // (H,)
    float*       __restrict__ out)      // (N, H) quantized values (as float)
{
    // Double-buffered async staging: 2 stages x (a-half + b-half) x 4 KiB = 16 KiB LDS
    __shared__ v4i sa[2][TPB];
    __shared__ v4i sb[2][TPB];

    const int   row  = blockIdx.x;      // one block per row
    const int   tid  = threadIdx.x;
    const float as_r = ascale[row];     // block-uniform -> scalar load

    const int* xa_base = x + (size_t)row * TWO_H;
    const int* xb_base = xa_base + HALF;

    // Issue one 16B-per-lane async copy of chunk `it` into stage `st`.
    // x is streamed exactly once -> non-temporal so the 256 MiB stream
    // doesn't cycle L2 and evict the hot scale/bias tables.
    auto issue = [&](int it, int st) {
        const int col = it * CHUNK + tid * VEC;
        uint64_t ga = (uint64_t)(uintptr_t)(xa_base + col);
        uint64_t gb = (uint64_t)(uintptr_t)(xb_base + col);
        uint32_t la = (uint32_t)(uintptr_t)&sa[st][tid];  // low 32 bits = LDS offset
        uint32_t lb = (uint32_t)(uintptr_t)&sb[st][tid];
        asm volatile(
            "global_load_async_to_lds_b128 %0, %1, off th:TH_LOAD_NT\n\t"
            "global_load_async_to_lds_b128 %2, %3, off th:TH_LOAD_NT"
            :: "v"(la), "v"(ga), "v"(lb), "v"(gb)
            : "memory");
    };

    issue(0, 0);

    #pragma unroll
    for (int it = 0; it < NCHUNK; ++it) {
        const int st = it & 1;
        if (it + 1 < NCHUNK) {
            // Close WAR window: prior DS reads of the buffer we are about to
            // overwrite must drain before the async engine may write it.
            asm volatile("s_wait_dscnt 0x0" ::: "memory");
            issue(it + 1, st ^ 1);
            // 4 async ops outstanding; wait until only the 2 newest remain.
            asm volatile("s_wait_asynccnt 0x2" ::: "memory");
        } else {
            asm volatile("s_wait_asynccnt 0x0" ::: "memory");
        }

        const v4i xa = sa[st][tid];   // ds_load_b128 from own slot (no barrier)
        const v4i xb = sb[st][tid];

        const int col = it * CHUNK + tid * VEC;
        const v4f wa = *(const v4f*)(wscale  + col);
        const v4f wb = *(const v4f*)(wscale  + HALF + col);
        const v4f ba = *(const v4f*)(bias    + col);
        const v4f bb = *(const v4f*)(bias    + HALF + col);
        const v4f qs = *(const v4f*)(qscale  + col);
        const v4f qo = *(const v4f*)(qoffset + col);

        v4f r;
        r.x = gate_quant((float)xa.x, (float)xb.x, wa.x, wb.x, as_r, ba.x, bb.x, qs.x, qo.x);
        r.y = gate_quant((float)xa.y, (float)xb.y, wa.y, wb.y, as_r, ba.y, bb.y, qs.y, qo.y);
        r.z = gate_quant((float)xa.z, (float)xb.z, wa.z, wb.z, as_r, ba.z, bb.z, qs.z, qo.z);
        r.w = gate_quant((float)xa.w, (float)xb.w, wa.w, wb.w, as_r, ba.w, bb.w, qs.w, qo.w);

        // Streamed output, never re-read -> non-temporal 128-bit store
        __builtin_nontemporal_store(r, (v4f*)(out + (size_t)row * HALF + col));
    }
}

// Second tuple element: quant_scale passthrough appended after y.
__global__ void copy_qscale_kernel(const float* __restrict__ qscale,
                                   float* __restrict__ out) {
    int i = blockIdx.x * blockDim.x + threadIdx.x;
    if (i < HALF) out[(size_t)N_ROWS * HALF + i] = qscale[i];
}

extern "C" void kernel_launch(void* const* d_in, const int* in_sizes, int n_in,
                              void* d_out, int out_size, void* d_ws, size_t ws_size,
                              hipStream_t stream) {
    (void)in_sizes; (void)n_in; (void)out_size; (void)d_ws; (void)ws_size;
    const int*   x  = (const int*)  d_in[0];
    const float* ws = (const float*)d_in[1];
    const float* as = (const float*)d_in[2];
    const float* bi = (const float*)d_in[3];
    const float* qs = (const float*)d_in[4];
    const float* qo = (const float*)d_in[5];
    float* out = (float*)d_out;

    swiglu_dequant_quant_kernel<<<dim3(N_ROWS), dim3(TPB), 0, stream>>>(
        x, ws, as, bi, qs, qo, out);
    copy_qscale_kernel<<<dim3((HALF + 255) / 256), dim3(256), 0, stream>>>(qs, out);
}